// EdgeEncoder_3685081940183
// MI455X (gfx1250) — compile-verified
//
#include <hip/hip_runtime.h>
#include <math.h>

// Problem constants (from reference setup_inputs)
#define B_   16
#define N_   100
#define L_   128
#define LE_  32
#define D_   768
#define OM_  4
#define OE_  4
#define P_   10
#define EPS_ 1e-8f

typedef __attribute__((ext_vector_type(2))) float v2f;
typedef __attribute__((ext_vector_type(8))) float v8f;

// Workspace layout (float offsets). Total = 61,520 floats = 246 KB.
#define WS_MENC   0                        // [B, D]   mention span means
#define WS_NMENC  (B_ * D_)                // [B]      ||m_enc||
#define WS_MOF    (WS_NMENC + B_)          // [B,Om,D] mention object P-means
#define WS_NM     (WS_MOF + B_ * OM_ * D_) // [B,Om]   ||mof||

// ---------------------------------------------------------------------------
// Kernel 1: mention-side prep (tiny). blocks 0..B-1 -> m_enc[b]; blocks
// B..B+B*Om-1 -> mof[b,i]. 256 threads, each owns 3 of the 768 dims.
// ---------------------------------------------------------------------------
__global__ __launch_bounds__(256)
void ee_prep_kernel(const float* __restrict__ mtf,    // [B,L,D]
                    const int*   __restrict__ startp, // [B]
                    const int*   __restrict__ endp,   // [B]
                    const float* __restrict__ mofeat, // [B,Om,P,D]
                    float*       __restrict__ ws) {
  __shared__ float red[256];
  const int tid  = threadIdx.x;
  const int part = blockIdx.x;

  float v0, v1, v2;
  float* dstVec;
  float* dstNorm;

  if (part < B_) {
    const int b = part;
    const int s = startp[b], e = endp[b];
    const float inv = 1.0f / (float)(e - s);   // end > start guaranteed
    const float* base = mtf + (size_t)b * L_ * D_;
    float a0 = 0.f, a1 = 0.f, a2 = 0.f;
    for (int l = s; l < e; ++l) {
      const float* row = base + (size_t)l * D_;
      a0 += row[tid]; a1 += row[tid + 256]; a2 += row[tid + 512];
    }
    v0 = a0 * inv; v1 = a1 * inv; v2 = a2 * inv;
    dstVec  = ws + WS_MENC + (size_t)b * D_;
    dstNorm = ws + WS_NMENC + b;
  } else {
    const int idx = part - B_;
    const int b = idx / OM_, i = idx - b * OM_;
    const float* base = mofeat + (size_t)(b * OM_ + i) * P_ * D_;
    float a0 = 0.f, a1 = 0.f, a2 = 0.f;
    for (int p = 0; p < P_; ++p) {
      const float* row = base + (size_t)p * D_;
      a0 += row[tid]; a1 += row[tid + 256]; a2 += row[tid + 512];
    }
    v0 = a0 * 0.1f; v1 = a1 * 0.1f; v2 = a2 * 0.1f;
    dstVec  = ws + WS_MOF + (size_t)(b * OM_ + i) * D_;
    dstNorm = ws + WS_NM + (b * OM_ + i);
  }

  dstVec[tid] = v0; dstVec[tid + 256] = v1; dstVec[tid + 512] = v2;

  red[tid] = v0 * v0 + v1 * v1 + v2 * v2;
  __syncthreads();
  for (int off = 128; off > 0; off >>= 1) {
    if (tid < off) red[tid] += red[tid + off];
    __syncthreads();
  }
  if (tid == 0) *dstNorm = sqrtf(red[0]);
}

// ---------------------------------------------------------------------------
// Kernel 2: one block per (b,n). Streams the 123 KB entity_object slice once
// (b128 loads, P-mean in registers), stages 10 vectors of length 768 in LDS,
// then computes ALL dot products (numerators + entity-side norms) as a single
// 16x16x768 f32 WMMA chain split over 8 waves.
//
//   A rows: 0=m_enc, 1..4=mof_i, 5=e_cls, 6..9=eof_j, 10..15=0 (zero row)
//   B cols: 0=e_cls, 1..4=eof_j, 5..15=0 (zero row)
//   C[0][0]      = m_enc . e_cls          (mtet numerator)
//   C[1+i][1+j]  = mof_i . eof_j          (miei numerators)
//   C[5][0]      = ||e_cls||^2
//   C[6+j][1+j]  = ||eof_j||^2
// ---------------------------------------------------------------------------
__global__ __launch_bounds__(256)
void ee_main_kernel(const float* __restrict__ etf,    // [B,N,Le,D]
                    const float* __restrict__ eofeat, // [B,N,Oe,P,D]
                    const float* __restrict__ mos,    // [B,Om]
                    const float* __restrict__ eos,    // [B,N,Oe]
                    const float* __restrict__ ws,
                    float*       __restrict__ out) {  // [2,B,N]
  constexpr int STR = D_ + 4;           // 772: 772 % 64 == 4 -> conflict-free
  __shared__ float xa[10][STR];         // staged vectors (A rows / B cols)
  __shared__ float zrow[STR];           // shared zero row for padding lanes
  __shared__ float cpart[8][8][32];     // per-wave partial C tiles
  __shared__ float cfull[16][16];       // reduced C

  const int tid = threadIdx.x;
  const int n   = blockIdx.x;
  const int b   = blockIdx.y;

  // ---- Phase 1: stage vectors into LDS --------------------------------
  for (int d = tid; d < STR; d += 256) zrow[d] = 0.f;
  // rows 0..4: m_enc and mof (from workspace, L2-hot)
  for (int idx = tid; idx < 5 * D_; idx += 256) {
    const int r = idx / D_, d = idx - r * D_;
    xa[r][d] = (r == 0) ? ws[WS_MENC + b * D_ + d]
                        : ws[WS_MOF + (b * OM_ + (r - 1)) * D_ + d];
  }
  // row 5: entity CLS token
  {
    const float* ecls = etf + (size_t)(b * N_ + n) * LE_ * D_;
    for (int d = tid; d < D_; d += 256) xa[5][d] = ecls[d];
  }
  // rows 6..9: entity object P-means (the 189 MB stream, b128 loads)
  {
    const float* base = eofeat + (size_t)(b * N_ + n) * OE_ * P_ * D_;
    for (int u = tid; u < OE_ * (D_ / 4); u += 256) {
      const int j = u / (D_ / 4);
      const int q = u - j * (D_ / 4);
      const float4* src = reinterpret_cast<const float4*>(base + (size_t)j * P_ * D_) + q;
      float4 acc = make_float4(0.f, 0.f, 0.f, 0.f);
      #pragma unroll
      for (int p = 0; p < P_; ++p) {
        const float4 v = src[p * (D_ / 4)];
        acc.x += v.x; acc.y += v.y; acc.z += v.z; acc.w += v.w;
      }
      acc.x *= 0.1f; acc.y *= 0.1f; acc.z *= 0.1f; acc.w *= 0.1f;
      *reinterpret_cast<float4*>(&xa[6 + j][q * 4]) = acc;   // 16B-aligned
    }
  }
  __syncthreads();

  // ---- Phase 2: WMMA chain, each wave owns a K-slice of 96 ------------
  const int wave = tid >> 5;
  const int lane = tid & 31;
  const int row  = lane & 15;   // A row / B col handled by this lane
  const int half = lane >> 4;

  // Branch-free fragment sourcing: padding lanes point at the zero row.
  // Pointer select happens once (v_cndmask), every ds_load in the loop is
  // unconditional; same-address reads across padding lanes broadcast.
  const float* arow = (row < 10) ? xa[row]     : zrow;
  const float* brow = (row < 5)  ? xa[5 + row] : zrow;
  const int ka = 2 * half;   // A f32 16x4 layout: lane half picks K pair
  const int kb = half;       // B: lane half h holds K = h and h+2 (not A's order)

  v8f c0 = {0.f, 0.f, 0.f, 0.f, 0.f, 0.f, 0.f, 0.f};
  v8f c1 = {0.f, 0.f, 0.f, 0.f, 0.f, 0.f, 0.f, 0.f};
  const int k0base = wave * (D_ / 8);          // 96-wide K slice per wave
  constexpr int NSTEP = (D_ / 8) / 4;          // 24 WMMA steps of K=4

  // Software-pipelined: load step s+1 fragments before WMMA of step s, and
  // alternate two accumulators to break the WMMA->WMMA RAW chain on C.
  v2f a_cur, b_cur;
  a_cur.x = arow[k0base + ka];     a_cur.y = arow[k0base + ka + 1];
  b_cur.x = brow[k0base + kb];     b_cur.y = brow[k0base + kb + 2];
  #pragma unroll
  for (int s = 0; s < NSTEP; ++s) {
    v2f a_nxt, b_nxt;
    if (s + 1 < NSTEP) {
      const int k0 = k0base + 4 * (s + 1);
      a_nxt.x = arow[k0 + ka]; a_nxt.y = arow[k0 + ka + 1];
      b_nxt.x = brow[k0 + kb]; b_nxt.y = brow[k0 + kb + 2];
    }
    if (s & 1) {
      c1 = __builtin_amdgcn_wmma_f32_16x16x4_f32(false, a_cur, false, b_cur,
                                                 (short)0, c1, false, false);
    } else {
      c0 = __builtin_amdgcn_wmma_f32_16x16x4_f32(false, a_cur, false, b_cur,
                                                 (short)0, c0, false, false);
    }
    a_cur = a_nxt; b_cur = b_nxt;
  }
  #pragma unroll
  for (int r = 0; r < 8; ++r) cpart[wave][r][lane] = c0[r] + c1[r];
  __syncthreads();

  // ---- Phase 3: reduce the 8 partial C tiles --------------------------
  {
    const int vg = tid >> 5;
    const int l  = tid & 31;
    float s = 0.f;
    #pragma unroll
    for (int w = 0; w < 8; ++w) s += cpart[w][vg][l];
    // C/D layout: VGPR r, lanes 0-15 -> M=r, lanes 16-31 -> M=r+8; N=l&15
    cfull[vg + ((l >= 16) ? 8 : 0)][l & 15] = s;
  }
  __syncthreads();

  // ---- Phase 4: finalize cosines + weighted mean ----------------------
  if (tid == 0) {
    const float n_menc = ws[WS_NMENC + b];
    const float n_ecls = sqrtf(fmaxf(cfull[5][0], 0.f));
    out[b * N_ + n] = cfull[0][0] / fmaxf(n_menc * n_ecls, EPS_);

    float ne[OE_];
    #pragma unroll
    for (int j = 0; j < OE_; ++j) ne[j] = sqrtf(fmaxf(cfull[6 + j][1 + j], 0.f));

    float ssum = 0.f, wsum = 0.f;
    #pragma unroll
    for (int i = 0; i < OM_; ++i) {
      const float nmi = ws[WS_NM + b * OM_ + i];
      const float msc = mos[b * OM_ + i];
      #pragma unroll
      for (int j = 0; j < OE_; ++j) {
        const float sim = cfull[1 + i][1 + j] / fmaxf(nmi * ne[j], EPS_);
        const float wgt = msc * eos[(b * N_ + n) * OE_ + j];
        ssum += sim * wgt;
        wsum += wgt;
      }
    }
    out[B_ * N_ + b * N_ + n] = ssum / wsum;
  }
}

// ---------------------------------------------------------------------------
extern "C" void kernel_launch(void* const* d_in, const int* in_sizes, int n_in,
                              void* d_out, int out_size, void* d_ws, size_t ws_size,
                              hipStream_t stream) {
  (void)in_sizes; (void)n_in; (void)out_size; (void)ws_size;
  const float* mtf    = (const float*)d_in[0]; // mention_text_feature [B,L,D]
  const int*   sp     = (const int*)  d_in[1]; // mention_start_pos [B]
  const int*   ep     = (const int*)  d_in[2]; // mention_end_pos [B]
  const float* mofeat = (const float*)d_in[3]; // mention_object_feature [B,Om,P,D]
  const float* mos    = (const float*)d_in[4]; // mention_object_score [B,Om]
  const float* etf    = (const float*)d_in[5]; // entity_text_feature [B,N,Le,D]
  const float* eofeat = (const float*)d_in[6]; // entity_object_feature [B,N,Oe,P,D]
  const float* eos    = (const float*)d_in[7]; // entity_object_score [B,N,Oe]
  float* out = (float*)d_out;
  float* ws  = (float*)d_ws;

  ee_prep_kernel<<<B_ * (1 + OM_), 256, 0, stream>>>(mtf, sp, ep, mofeat, ws);
  ee_main_kernel<<<dim3(N_, B_), 256, 0, stream>>>(etf, eofeat, mos, eos, ws, out);
}